// HashGridEncoding_4054449128092
// MI455X (gfx1250) — compile-verified
//
#include <hip/hip_runtime.h>

#ifndef __has_builtin
#define __has_builtin(x) 0
#endif

#define N_LEVELS 16
#define TBL_ENTRIES (1u << 19)   // 524288 entries per level
#define NPTS_BLOCK 256

typedef float vfloat4 __attribute__((ext_vector_type(4)));
typedef float vfloat2 __attribute__((ext_vector_type(2)));
typedef int   v4i     __attribute__((ext_vector_type(4)));

#define AS1 __attribute__((address_space(1)))
#define AS3 __attribute__((address_space(3)))

__global__ __launch_bounds__(NPTS_BLOCK)
void hashgrid_enc_kernel(const float* __restrict__ xs,
                         const float* __restrict__ tables,
                         float* __restrict__ out,
                         int npts)
{
    // floor(16 * 2^(i/3)) for i in 0..15
    constexpr float RESF[N_LEVELS] = {16.f, 20.f, 25.f, 32.f, 40.f, 50.f, 64.f, 80.f,
                                      101.f, 128.f, 161.f, 203.f, 256.f, 322.f, 406.f, 512.f};
    // min(res^3, 2^19)
    constexpr unsigned SIZE[N_LEVELS] = {4096u, 8000u, 15625u, 32768u, 64000u, 125000u,
                                         262144u, 512000u, 524288u, 524288u, 524288u,
                                         524288u, 524288u, 524288u, 524288u, 524288u};
    constexpr unsigned P1 = 2654435761u;
    constexpr unsigned P2 = 805459861u;

    // Level-0 table slice: 4096 entries x 2 floats = 32 KB, staged in LDS.
    __shared__ __align__(16) float lut0[4096 * 2];

    // ---- Stage level-0 table into LDS (async global->LDS on gfx1250) ----
    {
        const int tid = threadIdx.x;
#if defined(__gfx1250__) && __has_builtin(__builtin_amdgcn_global_load_async_to_lds_b128)
        for (int i = tid; i < 2048; i += NPTS_BLOCK) {
            AS1 v4i* g = (AS1 v4i*)tables + i;
            AS3 v4i* l = (AS3 v4i*)lut0 + i;
            __builtin_amdgcn_global_load_async_to_lds_b128(g, l, 0, 0);
        }
#if __has_builtin(__builtin_amdgcn_s_wait_asynccnt)
        __builtin_amdgcn_s_wait_asynccnt(0);
#else
        asm volatile("s_wait_asynccnt 0x0" ::: "memory");
#endif
#else
        const vfloat4* src = reinterpret_cast<const vfloat4*>(tables);
        vfloat4* dst = reinterpret_cast<vfloat4*>(lut0);
        for (int i = tid; i < 2048; i += NPTS_BLOCK) dst[i] = src[i];
#endif
        __syncthreads();
    }

    const int p = blockIdx.x * blockDim.x + threadIdx.x;
    if (p >= npts) return;

    const float* xp = xs + (size_t)3 * p;
    const float xv = xp[0];
    const float yv = xp[1];
    const float zv = xp[2];

    float acc[2 * N_LEVELS];

#pragma unroll
    for (int l = 0; l < N_LEVELS; ++l) {
        const float r = RESF[l];
        const float fx = xv * r;
        const float fy = yv * r;
        const float fz = zv * r;
        const int ix = (int)fx;   // trunc == floor for x >= 0
        const int iy = (int)fy;
        const int iz = (int)fz;
        const float cx = fx - (float)ix;
        const float cy = fy - (float)iy;
        const float cz = fz - (float)iz;

        // Pre-multiplied hash components: corner bit adds the prime.
        const unsigned hx0 = (unsigned)ix;          // prime 1
        const unsigned hx1 = hx0 + 1u;
        const unsigned hy0 = (unsigned)iy * P1;
        const unsigned hy1 = hy0 + P1;
        const unsigned hz0 = (unsigned)iz * P2;
        const unsigned hz1 = hz0 + P2;

        const float wx0 = 1.0f - cx;
        const float wy0 = 1.0f - cy;
        const float wz0 = 1.0f - cz;
        // (y,z) weight products, indexed by (k >> 1)
        const float wyz[4] = {wy0 * wz0, cy * wz0, wy0 * cz, cy * cz};

        const unsigned sz = SIZE[l];
        const float* tb = tables + (size_t)l * (TBL_ENTRIES * 2u);

        float a0 = 0.0f, a1 = 0.0f;
#pragma unroll
        for (int k = 0; k < 8; ++k) {
            const unsigned hv = ((k & 1) ? hx1 : hx0) ^
                                ((k & 2) ? hy1 : hy0) ^
                                ((k & 4) ? hz1 : hz0);
            const unsigned idx = hv % sz;   // sz is a compile-time constant per level
            const float wk = ((k & 1) ? cx : wx0) * wyz[k >> 1];
            vfloat2 f;
            if (l == 0) {
                f = reinterpret_cast<const vfloat2*>(lut0)[idx];       // ds_load_b64
            } else {
                f = *reinterpret_cast<const vfloat2*>(tb + 2u * idx); // global_load_b64
            }
            a0 = fmaf(wk, f.x, a0);
            a1 = fmaf(wk, f.y, a1);
        }
        acc[2 * l + 0] = a0;
        acc[2 * l + 1] = a1;
    }

    // 8 x b128 non-temporal stores: keep the 128MB output stream from
    // evicting the L2-resident hash tables (64MB < 192MB L2).
    vfloat4* dst4 = reinterpret_cast<vfloat4*>(out + (size_t)p * (2 * N_LEVELS));
#pragma unroll
    for (int i = 0; i < 8; ++i) {
        vfloat4 v = {acc[4 * i + 0], acc[4 * i + 1], acc[4 * i + 2], acc[4 * i + 3]};
        __builtin_nontemporal_store(v, dst4 + i);
    }
}

extern "C" void kernel_launch(void* const* d_in, const int* in_sizes, int n_in,
                              void* d_out, int out_size, void* d_ws, size_t ws_size,
                              hipStream_t stream) {
    const float* xs     = (const float*)d_in[0];  // [N, 3] float32
    const float* tables = (const float*)d_in[1];  // [16, 524288, 2] float32
    float* out          = (float*)d_out;          // [N, 32] float32

    const int npts = in_sizes[0] / 3;
    dim3 block(NPTS_BLOCK);
    dim3 grid((npts + NPTS_BLOCK - 1) / NPTS_BLOCK);
    hipLaunchKernelGGL(hashgrid_enc_kernel, grid, block, 0, stream,
                       xs, tables, out, npts);
}